// GATComplex_21930103013894
// MI455X (gfx1250) — compile-verified
//
#include <hip/hip_runtime.h>
#include <hip/hip_bf16.h>
#include <math.h>

// ---------------- CDNA5 WMMA types ----------------
typedef __attribute__((ext_vector_type(2))) float v2f;
typedef __attribute__((ext_vector_type(8))) float v8f;

#define N_NODES 50000
#define N_EDGES 800000
#define IN_DIM 64
#define HID 32
#define HEADS 4
#define HC 128          // HEADS*HID
#define NEG_SLOPE 0.2f

// ---------------- utility ----------------
__global__ void fill_kernel(float* __restrict__ p, int n, float v) {
    int t = blockIdx.x * blockDim.x + threadIdx.x;
    if (t < n) p[t] = v;
}

__device__ __forceinline__ void atomicMaxF(float* addr, float val) {
    // sign-split trick: works for mixed signs with -inf initialization
    if (val >= 0.0f) atomicMax((int*)addr, __float_as_int(val));
    else             atomicMin((unsigned int*)addr, __float_as_uint(val));
}

// ---------------- fp32 WMMA GEMM: C[M,NOUT] = act(A[M,K] @ W[NOUT,ldw](rows)^T + bias + extra) ----
// One wave (32 lanes) per 16-row tile; A fragments register-resident for the full K;
// W staged in LDS once per block (all 8 waves share B fragments).
// V_WMMA_F32_16X16X4_F32 layouts (ISA 7.12.2):
//   A 16x4 : lane<16 -> M=lane, K = 4*kk + vgpr + 2*(lane>=16)
//   B 4x16 : lane%16 -> N,      K = 4*kk + vgpr + 2*(lane>=16)
//   C 16x16: vgpr r : lanes0-15 -> M=r, lanes16-31 -> M=r+8; N = lane%16
template<int K, int NOUT, int ACT>
__global__ void wmma_gemm(const float* __restrict__ A,
                          const float* __restrict__ W, int ldw,
                          const float* __restrict__ bias,
                          const float* __restrict__ extra,
                          float* __restrict__ C, int M) {
    __shared__ float Ws[NOUT * K];          // max 64KB (conv2: 128x128 f32)

    // cooperative stage of W[:, :K] (row stride ldw) into contiguous LDS [NOUT][K]
    for (int idx = threadIdx.x; idx < NOUT * K; idx += blockDim.x) {
        const int row = idx / K, k = idx - row * K;     // K is power of 2 -> shifts
        Ws[idx] = W[(size_t)row * ldw + k];
    }
    __syncthreads();

    const int wave = threadIdx.x >> 5;
    const int lane = threadIdx.x & 31;
    const int rowTile = blockIdx.x * (blockDim.x >> 5) + wave;
    const int row0 = rowTile * 16;
    if (row0 >= M) return;                  // wave-uniform exit (after barrier!)
    const int hi = lane >> 4;               // 0 or 1
    const int lm = lane & 15;
    const int arow = row0 + lm;
    const bool rvalid = arow < M;

    // A fragments: one 64-bit base, constant offsets -> immediate-offset b64 loads
    const float* __restrict__ ap = A + (size_t)arow * K + 2 * hi;
    const v2f zero2 = {0.0f, 0.0f};
    v2f afrag[K / 4];
#pragma unroll
    for (int kk = 0; kk < K / 4; ++kk)
        afrag[kk] = rvalid ? *(const v2f*)(ap + 4 * kk) : zero2;

#pragma unroll
    for (int ot = 0; ot < NOUT / 16; ++ot) {
        const int col = ot * 16 + lm;
        const float* __restrict__ wp = &Ws[col * K + 2 * hi];   // LDS base per col tile
        v8f acc = {};
#pragma unroll
        for (int kk = 0; kk < K / 4; ++kk) {
            const v2f b = *(const v2f*)(wp + 4 * kk);           // ds_load_b64, imm offset
            acc = __builtin_amdgcn_wmma_f32_16x16x4_f32(
                false, afrag[kk], false, b, (short)0, acc, false, false);
        }
        float add = 0.0f;
        if (bias)  add += bias[col];
        if (extra) add += extra[col];
        float* __restrict__ cp = C + (size_t)row0 * NOUT + col;
#pragma unroll
        for (int r = 0; r < 8; ++r) {
            const int row = row0 + r + 8 * hi;
            if (row < M) {
                float v = acc[r] + add;
                if (ACT == 1) v = v > 0.0f ? v : 0.0f;   // relu
                cp[(size_t)(r + 8 * hi) * NOUT] = v;
            }
        }
    }
}

// ---------------- rates encoder + rcontrib = W2[:,128:160] @ r ----------------
__global__ void encode_rates(const float* __restrict__ rates,
                             const float* __restrict__ w1, const float* __restrict__ b1,
                             const float* __restrict__ w2, const float* __restrict__ b2,
                             const float* __restrict__ W2,
                             float* __restrict__ rcontrib) {
    __shared__ float mid[8];
    __shared__ float r[32];
    const int t = threadIdx.x;
    if (t < 8) {
        float s = b1[t];
        for (int k = 0; k < 16; ++k) s += w1[t * 16 + k] * rates[k];
        mid[t] = s > 0.0f ? s : 0.0f;
    }
    __syncthreads();
    if (t < 32) {
        float s = b2[t];
        for (int j = 0; j < 8; ++j) s += w2[t * 8 + j] * mid[j];
        r[t] = s;
    }
    __syncthreads();
    if (t < 128) {
        float s = 0.0f;
        for (int i = 0; i < 32; ++i) s += W2[t * 160 + 128 + i] * r[i];
        rcontrib[t] = s;
    }
}

// ---------------- per-(node,head) attention logits ----------------
__global__ void node_logits(const float* __restrict__ h,
                            const float* __restrict__ a_src,
                            const float* __restrict__ a_dst,
                            float* __restrict__ als, float* __restrict__ ald, int n) {
    int t = blockIdx.x * blockDim.x + threadIdx.x;
    if (t >= n * HEADS) return;
    const int node = t >> 2, hh = t & 3;
    float s0 = 0.0f, s1 = 0.0f;
    const float* hp = h + node * HC + hh * HID;
    const float* as = a_src + hh * HID;
    const float* ad = a_dst + hh * HID;
#pragma unroll 8
    for (int c = 0; c < HID; ++c) { float v = hp[c]; s0 += v * as[c]; s1 += v * ad[c]; }
    als[t] = s0; ald[t] = s1;
}

// ---------------- edge pass 1: segment max of leaky_relu logits ----------------
__global__ void edge_max(const int* __restrict__ ei,
                         const float* __restrict__ als, const float* __restrict__ ald,
                         float* __restrict__ mx) {
    int t = blockIdx.x * blockDim.x + threadIdx.x;
    const int total = (N_EDGES + N_NODES) * HEADS;
    if (t >= total) return;
    const int e = t >> 2, hh = t & 3;
    int s, d;
    if (e < N_EDGES) { s = ei[e]; d = ei[N_EDGES + e]; } else { s = d = e - N_EDGES; }
    float x = als[s * 4 + hh] + ald[d * 4 + hh];
    float l = x > 0.0f ? x : NEG_SLOPE * x;
    atomicMaxF(&mx[d * 4 + hh], l);
}

// ---------------- edge pass 2: e=exp(l-max); denom += e; accum[dst] += e*h[src] ----------------
// one wave per edge; lane = channel
__global__ void edge_scatter(const int* __restrict__ ei,
                             const float* __restrict__ h,
                             const float* __restrict__ als, const float* __restrict__ ald,
                             const float* __restrict__ mx,
                             float* __restrict__ denom, float* __restrict__ accum) {
    const int e = blockIdx.x * (blockDim.x >> 5) + (threadIdx.x >> 5);
    if (e >= N_EDGES + N_NODES) return;
    const int lane = threadIdx.x & 31;
    int s, d;
    if (e < N_EDGES) { s = ei[e]; d = ei[N_EDGES + e]; } else { s = d = e - N_EDGES; }
    float eh[HEADS];
#pragma unroll
    for (int hh = 0; hh < HEADS; ++hh) {
        float x = als[s * 4 + hh] + ald[d * 4 + hh];
        float l = x > 0.0f ? x : NEG_SLOPE * x;
        eh[hh] = expf(l - mx[d * 4 + hh]);
    }
    if (lane < 4) {
        float mine = lane == 0 ? eh[0] : lane == 1 ? eh[1] : lane == 2 ? eh[2] : eh[3];
        atomicAdd(&denom[d * 4 + lane], mine);
    }
#pragma unroll
    for (int hh = 0; hh < HEADS; ++hh) {
        atomicAdd(&accum[d * HC + hh * HID + lane],
                  h[s * HC + hh * HID + lane] * eh[hh]);
    }
}

// ---------------- normalize + bias + elu ----------------
__global__ void node_finish(const float* __restrict__ accum, const float* __restrict__ denom,
                            const float* __restrict__ bias, float* __restrict__ out, int n) {
    int t = blockIdx.x * blockDim.x + threadIdx.x;
    if (t >= n * HC) return;
    const int node = t >> 7, o = t & 127, hh = o >> 5;
    float v = accum[t] / (denom[node * 4 + hh] + 1e-16f) + bias[o];
    out[t] = v > 0.0f ? v : expm1f(v);      // elu(alpha=1)
}

// ---------------- deterministic mean pool ----------------
__global__ void mean_partial(const float* __restrict__ h3, float* __restrict__ part, int n) {
    const int c = threadIdx.x;              // 32 lanes = 32 cols
    const int b = blockIdx.x;
    float s = 0.0f;
    for (int node = b; node < n; node += gridDim.x) s += h3[node * HID + c];
    part[b * HID + c] = s;
}
__global__ void mean_final(const float* __restrict__ part, float* __restrict__ out,
                           int nblocks, float inv_n) {
    const int c = threadIdx.x;
    if (c >= HID) return;
    float s = 0.0f;
    for (int b = 0; b < nblocks; ++b) s += part[b * HID + c];
    out[c] = s * inv_n;
}

// ---------------- host launch ----------------
extern "C" void kernel_launch(void* const* d_in, const int* in_sizes, int n_in,
                              void* d_out, int out_size, void* d_ws, size_t ws_size,
                              hipStream_t stream) {
    const float* x       = (const float*)d_in[0];
    const int*   ei      = (const int*)  d_in[1];
    const float* rates   = (const float*)d_in[2];
    const float* W1      = (const float*)d_in[3];
    const float* a1_src  = (const float*)d_in[4];
    const float* a1_dst  = (const float*)d_in[5];
    const float* b1      = (const float*)d_in[6];
    const float* W2      = (const float*)d_in[7];
    const float* a2_src  = (const float*)d_in[8];
    const float* a2_dst  = (const float*)d_in[9];
    const float* b2      = (const float*)d_in[10];
    const float* enc_w1  = (const float*)d_in[11];
    const float* enc_b1  = (const float*)d_in[12];
    const float* enc_w2  = (const float*)d_in[13];
    const float* enc_b2  = (const float*)d_in[14];
    const float* fc_w    = (const float*)d_in[15];
    const float* fc_b    = (const float*)d_in[16];
    float* out = (float*)d_out;

    const int N = N_NODES, E2 = N_EDGES + N_NODES;
    const int RED_BLOCKS = 512;

    // workspace carve-up (floats)
    float* hbuf     = (float*)d_ws;             // [N*128] h-pre / h1a / accum2 target
    float* accbuf   = hbuf    + N * HC;         // [N*128] accum1 / h2-pre / h2a
    float* als      = accbuf  + N * HC;         // [N*4]
    float* ald      = als     + N * 4;          // [N*4]
    float* mx       = ald     + N * 4;          // [N*4]
    float* den      = mx      + N * 4;          // [N*4]
    float* h3       = den     + N * 4;          // [N*32]
    float* rcontrib = h3      + N * HID;        // [128]
    float* part     = rcontrib + HC;            // [512*32]

    const int TB = 256;
    const int gemm_blocks = ((N + 15) / 16 + 7) / 8;      // 8 waves per block
    const int nodeHeads   = (N * HEADS + TB - 1) / TB;
    const int nodeElems   = (N * HC + TB - 1) / TB;
    const int edgeHeads   = (E2 * HEADS + TB - 1) / TB;
    const int edgeWaves   = (E2 + 7) / 8;                 // 8 edges (waves) per 256-block
    const float NEG_INF = -INFINITY;

    // rates encoder + W2 residual contribution
    encode_rates<<<1, 128, 0, stream>>>(rates, enc_w1, enc_b1, enc_w2, enc_b2, W2, rcontrib);

    // ---- layer 1 ----
    wmma_gemm<IN_DIM, HC, 0><<<gemm_blocks, TB, 0, stream>>>(x, W1, IN_DIM, nullptr, nullptr, hbuf, N);
    node_logits<<<nodeHeads, TB, 0, stream>>>(hbuf, a1_src, a1_dst, als, ald, N);
    fill_kernel<<<nodeElems, TB, 0, stream>>>(accbuf, N * HC, 0.0f);
    fill_kernel<<<nodeHeads, TB, 0, stream>>>(mx,  N * 4, NEG_INF);
    fill_kernel<<<nodeHeads, TB, 0, stream>>>(den, N * 4, 0.0f);
    edge_max<<<edgeHeads, TB, 0, stream>>>(ei, als, ald, mx);
    edge_scatter<<<edgeWaves, TB, 0, stream>>>(ei, hbuf, als, ald, mx, den, accbuf);
    node_finish<<<nodeElems, TB, 0, stream>>>(accbuf, den, b1, hbuf, N);   // hbuf := h1a

    // ---- layer 2 (hcat folded: h1a @ W2[:, :128]^T + rcontrib) ----
    wmma_gemm<HC, HC, 0><<<gemm_blocks, TB, 0, stream>>>(hbuf, W2, 160, nullptr, rcontrib, accbuf, N);
    node_logits<<<nodeHeads, TB, 0, stream>>>(accbuf, a2_src, a2_dst, als, ald, N);
    fill_kernel<<<nodeElems, TB, 0, stream>>>(hbuf, N * HC, 0.0f);
    fill_kernel<<<nodeHeads, TB, 0, stream>>>(mx,  N * 4, NEG_INF);
    fill_kernel<<<nodeHeads, TB, 0, stream>>>(den, N * 4, 0.0f);
    edge_max<<<edgeHeads, TB, 0, stream>>>(ei, als, ald, mx);
    edge_scatter<<<edgeWaves, TB, 0, stream>>>(ei, accbuf, als, ald, mx, den, hbuf);
    node_finish<<<nodeElems, TB, 0, stream>>>(hbuf, den, b2, accbuf, N);   // accbuf := h2a

    // ---- fc + relu (WMMA) then deterministic mean pool ----
    wmma_gemm<HC, HID, 1><<<gemm_blocks, TB, 0, stream>>>(accbuf, fc_w, HC, fc_b, nullptr, h3, N);
    mean_partial<<<RED_BLOCKS, 32, 0, stream>>>(h3, part, N);
    mean_final<<<1, 32, 0, stream>>>(part, out, RED_BLOCKS, 1.0f / (float)N);

    (void)in_sizes; (void)n_in; (void)out_size; (void)ws_size;
}